// SelfAttentionLayer_65171833750241
// MI455X (gfx1250) — compile-verified
//
#include <hip/hip_runtime.h>
#include <math.h>

// MI455X plan: bandwidth-bound (~400 MB moved, ~17us floor @ 23.3 TB/s).
// x (134MB) fits the 192MB L2 -> keep it resident across the 3 passes;
// output written with non-temporal stores so the 134MB write doesn't evict x.
// scores pass uses V_WMMA_F32_16X16X4_F32 (full f32 precision matvec).

typedef float v2f __attribute__((ext_vector_type(2)));
typedef float v4f __attribute__((ext_vector_type(4)));
typedef float v8f __attribute__((ext_vector_type(8)));

constexpr int Bn = 4;
constexpr int Sn = 8192;
constexpr int Dn = 1024;
constexpr int SBLK = 64;            // #partial chunks along S
constexpr int SCH  = Sn / SBLK;     // 128 rows per chunk
constexpr float INV_SQRT_D = 0.03125f;  // 1/sqrt(1024)

// ---- Pass 1a: deterministic partial column sums: partials[b][sblk][d] ----
__global__ void colsum_partial_k(const float* __restrict__ x,
                                 float* __restrict__ partials) {
    const int tid  = threadIdx.x;
    const unsigned blk = blockIdx.x;          // B * SBLK * (D/256) = 1024 blocks
    const int dblk = blk & 3;                 // D/256 = 4
    const int sblk = (blk >> 2) & 63;         // SBLK = 64
    const int b    = blk >> 8;
    const int d    = dblk * 256 + tid;
    const float* p = x + (size_t)(b * Sn + sblk * SCH) * Dn + d;
    float acc = 0.f;
#pragma unroll 8
    for (int i = 0; i < SCH; ++i) acc += p[(size_t)i * Dn];
    partials[(size_t)(b * SBLK + sblk) * Dn + d] = acc;
}

// ---- Pass 1b: reduce partials -> x_sum * (1/sqrt(D)), fixed order ----
__global__ void colsum_final_k(const float* __restrict__ partials,
                               float* __restrict__ xsum) {
    const int idx = blockIdx.x * 256 + threadIdx.x;   // 0..B*D-1
    const int b = idx >> 10;
    const int d = idx & 1023;
    const float* p = partials + (size_t)b * SBLK * Dn + d;
    float acc = 0.f;
#pragma unroll 8
    for (int i = 0; i < SBLK; ++i) acc += p[(size_t)i * Dn];
    xsum[idx] = acc * INV_SQRT_D;
}

// ---- Pass 2: scores[b][s] = x[b,s,:] . xsum_scaled[b,:] via f32 WMMA ----
// One wave computes 16 sequence rows. A = 16x4 tile of x, B = xsum broadcast
// across the 16 columns, accumulated over D in steps of 4.
// A layout (ISA 7.12.2): lane m (0-15): v0=A[m][0], v1=A[m][1];
//                        lane m+16:     v0=A[m][2], v1=A[m][3].
// B layout (dual):       lane n (0-15): v0=B[0][n], v1=B[1][n];
//                        lane n+16:     v0=B[2][n], v1=B[3][n].
// Broadcast B[k][n]=xsum[d+k] => per-lane float2 at xsum + d + 2*(lane>>4).
__global__ void scores_wmma_k(const float* __restrict__ x,
                              const float* __restrict__ xsum,
                              float* __restrict__ scores) {
    const int lane  = threadIdx.x & 31;
    const int wave  = threadIdx.x >> 5;
    const int group = blockIdx.x * 8 + wave;      // 8 waves / block
    const int b     = group >> 9;                 // S/16 = 512 groups per batch
    const int s0    = (group & 511) << 4;
    const int hi    = lane >> 4;

    const float* arow = x + (size_t)(b * Sn + s0 + (lane & 15)) * Dn + 2 * hi;
    const float* brow = xsum + b * Dn + 2 * hi;

    v8f c = {};
#pragma unroll 4
    for (int d = 0; d < Dn; d += 4) {
        v2f a  = *(const v2f*)(arow + d);
        v2f bb = *(const v2f*)(brow + d);
        c = __builtin_amdgcn_wmma_f32_16x16x4_f32(
                /*neg_a=*/false, a, /*neg_b=*/false, bb,
                /*c_mod=*/(short)0, c, /*reuse_a=*/false, /*reuse_b=*/false);
    }
    // D layout: VGPR j -> M=j (lanes 0-15), M=j+8 (lanes 16-31); all N equal.
    if ((lane & 15) == 0) {
        float* sp = scores + (size_t)b * Sn + s0 + 8 * hi;
#pragma unroll
        for (int j = 0; j < 8; ++j) sp[j] = c[j];
    }
}

// ---- Pass 3: softmax over S per batch, in place (scores -> attn) ----
__global__ void softmax_k(float* __restrict__ scores) {
    __shared__ float red[1024];
    const int tid = threadIdx.x;
    float* sp = scores + (size_t)blockIdx.x * Sn;

    float v[8];
    float mx = -3.4e38f;
#pragma unroll
    for (int i = 0; i < 8; ++i) {
        v[i] = sp[i * 1024 + tid];
        mx = fmaxf(mx, v[i]);
    }
    red[tid] = mx;
    __syncthreads();
    for (int off = 512; off > 0; off >>= 1) {
        if (tid < off) red[tid] = fmaxf(red[tid], red[tid + off]);
        __syncthreads();
    }
    mx = red[0];
    __syncthreads();

    float e[8];
    float sum = 0.f;
#pragma unroll
    for (int i = 0; i < 8; ++i) {
        e[i] = expf(v[i] - mx);
        sum += e[i];
    }
    red[tid] = sum;
    __syncthreads();
    for (int off = 512; off > 0; off >>= 1) {
        if (tid < off) red[tid] += red[tid + off];
        __syncthreads();
    }
    const float inv = 1.f / red[0];
#pragma unroll
    for (int i = 0; i < 8; ++i) sp[i * 1024 + tid] = e[i] * inv;
}

// ---- Pass 4: out = w_v * x * attn, vectorized, non-temporal stores ----
// One block per (b,s) row: 256 threads x float4 = 1024 floats.
__global__ void out_k(const float* __restrict__ x,
                      const float* __restrict__ wv,
                      const float* __restrict__ attn,
                      float* __restrict__ out) {
    const size_t row = blockIdx.x;     // b*S + s
    const int tid = threadIdx.x;
    const float a = attn[row];
    const v4f xv = ((const v4f*)(x + row * Dn))[tid];
    const v4f wvv = ((const v4f*)wv)[tid];
    v4f o = xv * wvv * a;
    __builtin_nontemporal_store(o, ((v4f*)(out + row * Dn)) + tid);
}

extern "C" void kernel_launch(void* const* d_in, const int* in_sizes, int n_in,
                              void* d_out, int out_size, void* d_ws, size_t ws_size,
                              hipStream_t stream) {
    const float* x   = (const float*)d_in[0];
    // d_in[1] = w_q, d_in[2] = w_k : dead in the reference graph.
    const float* w_v = (const float*)d_in[3];
    float* out = (float*)d_out;

    float* ws = (float*)d_ws;
    float* partials = ws;                         // B*SBLK*D = 262144 floats
    float* xsum     = ws + (size_t)Bn * SBLK * Dn;        // B*D   = 4096
    float* scores   = xsum + (size_t)Bn * Dn;             // B*S   = 32768 (becomes attn)

    colsum_partial_k<<<Bn * SBLK * (Dn / 256), 256, 0, stream>>>(x, partials);
    colsum_final_k<<<(Bn * Dn) / 256, 256, 0, stream>>>(partials, xsum);
    scores_wmma_k<<<(Bn * Sn / 16) / 8, 256, 0, stream>>>(x, xsum, scores);
    softmax_k<<<Bn, 1024, 0, stream>>>(scores);
    out_k<<<Bn * Sn, 256, 0, stream>>>(x, w_v, scores, out);
}